// BCWildfireSTGNN_17437567222539
// MI455X (gfx1250) — compile-verified
//
#include <hip/hip_runtime.h>
#include <cstddef>
#include <cstdint>

typedef __attribute__((ext_vector_type(2))) float v2f;
typedef __attribute__((ext_vector_type(8))) float v8f;
typedef __attribute__((ext_vector_type(4))) unsigned int v4u;
typedef __attribute__((ext_vector_type(8))) int v8i_t;
typedef __attribute__((ext_vector_type(4))) int v4i_t;

#define B_    8
#define T_    24
#define N_    2000
#define FIN_  16
#define H_    64
#define E_    16000
#define DH_   16
#define ROWS_ (B_*T_*N_)   /* 384000 */
#define M_    (B_*N_)      /* 16000  */
#define EPS_  1e-5f

#if __has_builtin(__builtin_amdgcn_tensor_load_to_lds) && __has_builtin(__builtin_amdgcn_s_wait_tensorcnt)
#define HAVE_TDM 1
#else
#define HAVE_TDM 0
#endif

// ---------------------------------------------------------------------------
// TDM: DMA a 2D tile (rows x cols f32, global row stride in elements) into
// LDS, packed row-major. D# per CDNA5 ISA 8.3/8.4. Issue from ONE wave, then
// s_wait_tensorcnt(0) in that wave + __syncthreads() for the workgroup.
// This toolchain uses the 6-arg builtin: (g0 v4u, g1 v8i, g2 v4i, g3 v4i,
// extra v8i, cpol i32); groups 2/3 unused for 2D tiles -> zero.
// ---------------------------------------------------------------------------
#if HAVE_TDM
__device__ __forceinline__ void tdm_load_tile(const float* gsrc, unsigned lds_off_bytes,
                                              int rows, int cols, long long row_stride_elems)
{
  unsigned long long ga = (unsigned long long)(uintptr_t)gsrc;
  unsigned long long s0 = (unsigned long long)row_stride_elems;
  v4u g0;
  g0[0] = 1u;                                           // count=1 (valid user D#)
  g0[1] = lds_off_bytes;                                // lds_addr (bytes)
  g0[2] = (unsigned)(ga & 0xffffffffu);                 // global_addr[31:0]
  g0[3] = (unsigned)((ga >> 32) & 0x1ffffffu) | (2u << 30); // global_addr[56:32] | type=2
  v8i_t g1;
  g1[0] = (int)(2u << 16);                              // data_size=4B, mask=0, no flags
  g1[1] = (int)(((unsigned)cols & 0xffffu) << 16);      // tensor_dim0[15:0]
  g1[2] = (int)((((unsigned)cols >> 16) & 0xffffu) |
                (((unsigned)rows & 0xffffu) << 16));    // tensor_dim0[31:16] | tensor_dim1[15:0]
  g1[3] = (int)((((unsigned)rows >> 16) & 0xffffu) |
                (((unsigned)cols & 0xffffu) << 16));    // tensor_dim1[31:16] | tile_dim0
  g1[4] = (int)((unsigned)rows & 0xffffu);              // tile_dim1 | tile_dim2=0
  g1[5] = (int)(unsigned)(s0 & 0xffffffffu);            // tensor_dim0_stride[31:0]
  g1[6] = (int)(unsigned)((s0 >> 32) & 0xffffu);        // stride[47:32] | dim1_stride[15:0]=0
  g1[7] = 0;
  v4i_t z4 = {0, 0, 0, 0};
  v8i_t z8 = {0, 0, 0, 0, 0, 0, 0, 0};
  __builtin_amdgcn_tensor_load_to_lds(g0, g1, z4, z4, z8, 0);
}
#endif

__device__ __forceinline__ void stage_tile_wave0(const float* gsrc, float* lds_dst,
                                                 int rows, int cols, long long stride)
{
#if HAVE_TDM
  tdm_load_tile(gsrc, (unsigned)(uintptr_t)lds_dst, rows, cols, stride);
#else
  int lane = threadIdx.x & 31;
  for (int idx = lane; idx < rows * cols; idx += 32)
    lds_dst[idx] = gsrc[(size_t)(idx / cols) * stride + (idx % cols)];
#endif
}

__device__ __forceinline__ void stage_wait_wave0(void)
{
#if HAVE_TDM
  __builtin_amdgcn_s_wait_tensorcnt(0);
#endif
}

// ---------------------------------------------------------------------------
// fp32 WMMA 16x16 tile helpers (V_WMMA_F32_16X16X4_F32, wave32)
// A: 16 rows (M = lane%16), K = k0 + 2*(lane/16) + {0,1}
// B: B[k][n] at Bm[k*bk + n*bn], n = n0 + lane%16, K mirrored like A
// C/D: row = v + 8*(lane/16), col = lane%16
// ---------------------------------------------------------------------------
__device__ __forceinline__ void wmma_accum(const float* __restrict__ A, int a_stride,
                                           const float* __restrict__ Bm, int bk, int bn,
                                           int n0, int K, v8f& c)
{
  int lane = threadIdx.x & 31;
  int half = lane >> 4;
  int l16  = lane & 15;
  int ncol = n0 + l16;
  for (int k0 = 0; k0 < K; k0 += 4) {
    int ka = k0 + 2 * half;
    v2f a, b;
    a.x = A[(size_t)l16 * a_stride + ka];
    a.y = A[(size_t)l16 * a_stride + ka + 1];
    b.x = Bm[(size_t)ka * bk + (size_t)ncol * bn];
    b.y = Bm[(size_t)(ka + 1) * bk + (size_t)ncol * bn];
    c = __builtin_amdgcn_wmma_f32_16x16x4_f32(false, a, false, b, (short)0, c, false, false);
  }
}

__device__ __forceinline__ void wmma_store(float* __restrict__ C, int c_stride,
                                           const v8f& c, int relu)
{
  int lane = threadIdx.x & 31;
  int half = lane >> 4;
  int l16  = lane & 15;
#pragma unroll
  for (int v = 0; v < 8; ++v) {
    float x = c[v];
    if (relu) x = fmaxf(x, 0.f);
    C[(size_t)(v + 8 * half) * c_stride + l16] = x;
  }
}

// ---------------------------------------------------------------------------
// Generic GEMM: C[M x Nc] = act(A @ B + bias). One block = 64 rows x 64 cols:
// A block tile (64 x K) staged to LDS by TDM once, 4 waves each compute a
// 16-col tile for 4 row tiles (16 WMMA C-tiles per block).
// grid.x = M/64, grid.y = Nc/64.
// ---------------------------------------------------------------------------
__global__ void gemm_kernel(const float* __restrict__ A, long long a_off, int a_stride,
                            const float* __restrict__ Bm, int bk, int bn,
                            const float* __restrict__ bias,
                            float* __restrict__ C, long long c_off, int c_stride,
                            int K, int relu)
{
  __shared__ float As[64 * 64];                      // max K = 64
  int wave = threadIdx.x >> 5;
  int lane = threadIdx.x & 31;
  int l16  = lane & 15;
  const float* Ablk = A + a_off + (size_t)blockIdx.x * 64 * a_stride;
  if (wave == 0) {
    stage_tile_wave0(Ablk, As, 64, K, a_stride);
    stage_wait_wave0();
  }
  __syncthreads();
  int n0 = (blockIdx.y * 4 + wave) * 16;
  float bv = bias ? bias[n0 + l16] : 0.f;
  for (int rt = 0; rt < 4; ++rt) {
    v8f c;
#pragma unroll
    for (int v = 0; v < 8; ++v) c[v] = bv;
    wmma_accum(As + rt * 16 * K, K, Bm, bk, bn, n0, K, c);
    float* Ct = C + c_off + ((size_t)blockIdx.x * 64 + rt * 16) * c_stride + n0;
    wmma_store(Ct, c_stride, c, relu);
  }
}

// ---------------------------------------------------------------------------
// conv1x3 along T in (B,T,N,H) layout: 3 shifted GEMMs with W[co][ci][0][dt].
// grid (B*T, N/16), block 128: wave w owns output-channel tile co0 = w*16.
// The 3 shifted 16x64 input tiles are TDM-staged into LDS.
// ---------------------------------------------------------------------------
__global__ void conv1x3_kernel(const float* __restrict__ h, const float* __restrict__ w,
                               const float* __restrict__ bias, float* __restrict__ y, int relu)
{
  __shared__ float As[3 * 16 * 64];
  int bt = blockIdx.x;
  int b  = bt / T_, t = bt % T_;
  int n0 = blockIdx.y * 16;
  int wave = threadIdx.x >> 5;
  int lane = threadIdx.x & 31;
  int l16  = lane & 15;
  int co0  = wave * 16;
  if (wave == 0) {
    for (int dt = 0; dt < 3; ++dt) {
      int ts = t + dt - 1;
      if (ts < 0 || ts >= T_) continue;
      stage_tile_wave0(h + ((((size_t)b) * T_ + ts) * N_ + n0) * H_,
                       As + dt * 16 * 64, 16, 64, 64);
    }
    stage_wait_wave0();
  }
  __syncthreads();
  v8f c;
  float bv = bias[co0 + l16];
#pragma unroll
  for (int v = 0; v < 8; ++v) c[v] = bv;
  for (int dt = 0; dt < 3; ++dt) {
    int ts = t + dt - 1;
    if (ts < 0 || ts >= T_) continue;                 // uniform per block
    // B[k=ci][n=co] = w[(co*H_ + ci)*3 + dt]  -> bk=3, bn=H_*3
    wmma_accum(As + dt * 16 * 64, H_, w + dt, 3, H_ * 3, co0, H_, c);
  }
  float* C = y + (((size_t)bt) * N_ + n0) * H_ + co0;
  wmma_store(C, H_, c, relu);
}

// --------------------------- small utility kernels -------------------------
__global__ void fill_kernel(float* p, float v, size_t n) {
  size_t i = (size_t)blockIdx.x * blockDim.x + threadIdx.x;
  if (i < n) p[i] = v;
}
__global__ void fill_bias_kernel(float* p, const float* __restrict__ bias, size_t n) {
  size_t i = (size_t)blockIdx.x * blockDim.x + threadIdx.x;
  if (i < n) p[i] = bias[i & (H_ - 1)];
}
__global__ void deg_accum_kernel(const int* __restrict__ ei, const float* __restrict__ ew,
                                 float* deg) {
  int e = blockIdx.x * blockDim.x + threadIdx.x;
  if (e < E_) atomicAdd(&deg[ei[E_ + e]], ew[e]);
}
__global__ void dinv_kernel(const float* __restrict__ deg, float* dinv) {
  int n = blockIdx.x * blockDim.x + threadIdx.x;
  if (n < N_) { float d = deg[n]; dinv[n] = d > 0.f ? rsqrtf(d) : 0.f; }
}
__global__ void norm_kernel(const int* __restrict__ ei, const float* __restrict__ ew,
                            const float* __restrict__ dinv, float* nrm) {
  int e = blockIdx.x * blockDim.x + threadIdx.x;
  if (e < E_) nrm[e] = dinv[ei[e]] * ew[e] * dinv[ei[E_ + e]];
}

// GCN scatter: out[bt,dst,:] += xl[bt,src,:]*norm  (grid (E+N, B*T), block 64)
__global__ void gcn_scatter_kernel(const float* __restrict__ xl, const int* __restrict__ ei,
                                   const float* __restrict__ nrm, const float* __restrict__ dinv,
                                   float* __restrict__ out)
{
  int e = blockIdx.x, bt = blockIdx.y, k = threadIdx.x;
  int s, d; float w;
  if (e < E_) { s = ei[e]; d = ei[E_ + e]; w = nrm[e]; }
  else        { s = d = e - E_; float di = dinv[s]; w = di * di; }
  float v = xl[(((size_t)bt) * N_ + s) * H_ + k] * w;
  atomicAdd(&out[(((size_t)bt) * N_ + d) * H_ + k], v);
}

// LayerNorm over last dim (64), one wave per row, optional ReLU
__global__ void ln_kernel(const float* __restrict__ in, const float* __restrict__ g,
                          const float* __restrict__ bvec, float* __restrict__ out,
                          int relu, size_t rows)
{
  size_t row = (size_t)blockIdx.x * (blockDim.x / 32) + (threadIdx.x >> 5);
  if (row >= rows) return;
  int lane = threadIdx.x & 31;
  const float* r = in + row * H_;
  float x0 = r[lane], x1 = r[lane + 32];
  float s = x0 + x1;
  for (int off = 16; off; off >>= 1) s += __shfl_xor(s, off, 32);
  float mean = s * (1.f / H_);
  float d0 = x0 - mean, d1 = x1 - mean;
  float vs = d0 * d0 + d1 * d1;
  for (int off = 16; off; off >>= 1) vs += __shfl_xor(vs, off, 32);
  float rstd = rsqrtf(vs * (1.f / H_) + EPS_);
  float y0 = d0 * rstd * g[lane] + bvec[lane];
  float y1 = d1 * rstd * g[lane + 32] + bvec[lane + 32];
  if (relu) { y0 = fmaxf(y0, 0.f); y1 = fmaxf(y1, 0.f); }
  float* o = out + row * H_;
  o[lane] = y0; o[lane + 32] = y1;
}

// BatchNorm stats over all rows per channel (sum, sumsq) into stats[128]
__global__ void bn_stats_kernel(const float* __restrict__ y, const float* __restrict__ res,
                                float* __restrict__ stats)
{
  int c = threadIdx.x & (H_ - 1);
  int sub = threadIdx.x >> 6;  // 0..3
  float s = 0.f, q = 0.f;
  for (size_t r = (size_t)blockIdx.x * 4 + sub; r < (size_t)ROWS_; r += (size_t)gridDim.x * 4) {
    float v = y[r * H_ + c] + res[r * H_ + c];
    s += v; q += v * v;
  }
  __shared__ float ls[256], lq[256];
  ls[threadIdx.x] = s; lq[threadIdx.x] = q;
  __syncthreads();
  if (threadIdx.x < 64) {
    s = ls[threadIdx.x] + ls[threadIdx.x + 64] + ls[threadIdx.x + 128] + ls[threadIdx.x + 192];
    q = lq[threadIdx.x] + lq[threadIdx.x + 64] + lq[threadIdx.x + 128] + lq[threadIdx.x + 192];
    atomicAdd(&stats[c], s);
    atomicAdd(&stats[64 + c], q);
  }
}

__global__ void bn_apply_kernel(const float* __restrict__ y, float* __restrict__ res_inout,
                                const float* __restrict__ stats, const float* __restrict__ g,
                                const float* __restrict__ bvec)
{
  size_t i = (size_t)blockIdx.x * blockDim.x + threadIdx.x;
  if (i >= (size_t)ROWS_ * H_) return;
  int c = (int)(i & (H_ - 1));
  float mean = stats[c] * (1.f / ROWS_);
  float var  = stats[64 + c] * (1.f / ROWS_) - mean * mean;
  float v = y[i] + res_inout[i];
  res_inout[i] = (v - mean) * rsqrtf(var + EPS_) * g[c] + bvec[c];
}

// ---------------------------------------------------------------------------
// Fused MHA + residual + LayerNorm per sequence (one block per m=(b,n)).
// The 24x64 strided sequence gather is a TDM 2D tile (row stride N*H*4 B).
// QKV and output projection via fp32 WMMA on LDS tiles (zero-padded rows).
// ---------------------------------------------------------------------------
__global__ void mha_kernel(const float* __restrict__ h,
                           const float* __restrict__ in_w, const float* __restrict__ in_b,
                           const float* __restrict__ out_w, const float* __restrict__ out_b,
                           const float* __restrict__ ln_g, const float* __restrict__ ln_b,
                           float* __restrict__ outb)
{
  __shared__ float hn_s[32 * 64];
  __shared__ float qkv_s[32 * 192];
  __shared__ float o_s[32 * 64];
  __shared__ float p_s[4][32];
  int m = blockIdx.x;
  int b = m / N_, n = m % N_;
  int tid = threadIdx.x;
  int wave = tid >> 5, lane = tid & 31;

#if HAVE_TDM
  if (wave == 0)
    stage_tile_wave0(h + ((((size_t)b) * T_) * N_ + n) * H_, hn_s, T_, H_,
                     (long long)N_ * H_);
  for (int i = tid; i < 32 * 64; i += 128) {
    if (i >= T_ * 64) hn_s[i] = 0.f;   // zero pad rows 24..31 (disjoint from DMA)
    o_s[i] = 0.f;
  }
  if (wave == 0) stage_wait_wave0();
#else
  for (int i = tid; i < 32 * 64; i += 128) {
    int r = i >> 6, k = i & 63;
    hn_s[i] = (r < T_) ? h[((((size_t)b) * T_ + r) * N_ + n) * H_ + k] : 0.f;
    o_s[i] = 0.f;
  }
#endif
  __syncthreads();

  // QKV = hn(24x64, padded to 32) @ in_w^T(64x192) + in_b : 2x12 tiles
  for (int tile = wave; tile < 24; tile += 4) {
    int rt = tile / 12, ct = tile % 12;
    int l16 = lane & 15;
    v8f c; float bv = in_b[ct * 16 + l16];
#pragma unroll
    for (int v = 0; v < 8; ++v) c[v] = bv;
    wmma_accum(hn_s + rt * 16 * 64, 64, in_w, 1, 64, ct * 16, 64, c);
    wmma_store(qkv_s + rt * 16 * 192 + ct * 16, 192, c, 0);
  }
  __syncthreads();

  // attention: head = wave; q|k|v at col offsets 0|64|128
  int hd = wave;
  for (int t1 = 0; t1 < T_; ++t1) {
    int j = lane < T_ ? lane : T_ - 1;
    float s = 0.f;
    for (int d = 0; d < DH_; ++d)
      s += qkv_s[t1 * 192 + hd * 16 + d] * qkv_s[j * 192 + 64 + hd * 16 + d];
    s *= 0.25f;  // 1/sqrt(16)
    float sm = (lane < T_) ? s : -3.4e38f;
    float mx = sm;
    for (int off = 16; off; off >>= 1) mx = fmaxf(mx, __shfl_xor(mx, off, 32));
    float e = (lane < T_) ? __expf(sm - mx) : 0.f;
    float den = e;
    for (int off = 16; off; off >>= 1) den += __shfl_xor(den, off, 32);
    p_s[hd][lane] = e / den;
    __syncthreads();
    if (lane < DH_) {
      float acc = 0.f;
      for (int j2 = 0; j2 < T_; ++j2)
        acc += p_s[hd][j2] * qkv_s[j2 * 192 + 128 + hd * 16 + lane];
      o_s[t1 * 64 + hd * 16 + lane] = acc;
    }
    __syncthreads();
  }

  // out projection + residual -> r_s (reuse qkv_s memory)
  float* r_s = qkv_s;
  for (int tile = wave; tile < 8; tile += 4) {
    int rt = tile >> 2, ct = tile & 3;
    int l16 = lane & 15, half = lane >> 4;
    v8f c; float bv = out_b[ct * 16 + l16];
#pragma unroll
    for (int v = 0; v < 8; ++v) c[v] = bv;
    wmma_accum(o_s + rt * 16 * 64, 64, out_w, 1, 64, ct * 16, 64, c);
#pragma unroll
    for (int v = 0; v < 8; ++v) {
      int rr = rt * 16 + v + 8 * half;
      int cc = ct * 16 + l16;
      r_s[rr * 64 + cc] = c[v] + hn_s[rr * 64 + cc];
    }
  }
  __syncthreads();

  // LayerNorm rows 0..23 -> outb in (B*N, T, H) layout
  if (tid < T_) {
    int row = tid;
    float mean = 0.f;
    for (int k = 0; k < H_; ++k) mean += r_s[row * 64 + k];
    mean *= (1.f / H_);
    float var = 0.f;
    for (int k = 0; k < H_; ++k) { float d = r_s[row * 64 + k] - mean; var += d * d; }
    var *= (1.f / H_);
    float rstd = rsqrtf(var + EPS_);
    for (int k = 0; k < H_; ++k)
      outb[(((size_t)m) * T_ + row) * H_ + k] =
          (r_s[row * 64 + k] - mean) * rstd * ln_g[k] + ln_b[k];
  }
}

// --------------------------- GRU gate elementwise --------------------------
__device__ __forceinline__ float sigf(float x) { return 1.f / (1.f + __expf(-x)); }

__global__ void gru_gate_kernel(const float* __restrict__ gi, const float* __restrict__ gh,
                                const float* __restrict__ hprev, float* __restrict__ hnew,
                                float* __restrict__ ys, long long ys_off, int ys_stride)
{
  size_t i = (size_t)blockIdx.x * blockDim.x + threadIdx.x;
  if (i >= (size_t)M_ * H_) return;
  size_t mrow = i >> 6;
  int k = (int)(i & 63);
  const float* gir = gi + mrow * 192;
  const float* ghr = gh + mrow * 192;
  float r = sigf(gir[k] + ghr[k]);
  float z = sigf(gir[64 + k] + ghr[64 + k]);
  float nn = tanhf(gir[128 + k] + r * ghr[128 + k]);
  float hv = (1.f - z) * nn + z * hprev[i];
  hnew[i] = hv;
  if (ys) ys[ys_off + mrow * ys_stride + k] = hv;
}

// head: relu(last@fc1 + b1) @ fc2 + b2; one wave per row
__global__ void fc_kernel(const float* __restrict__ last, const float* __restrict__ w1,
                          const float* __restrict__ b1, const float* __restrict__ w2,
                          const float* __restrict__ b2, float* __restrict__ out)
{
  int m = blockIdx.x * (blockDim.x / 32) + (threadIdx.x >> 5);
  if (m >= M_) return;
  int lane = threadIdx.x & 31;
  const float* r = last + (size_t)m * H_;
  float f = b1[lane];
  for (int k = 0; k < H_; ++k) f += r[k] * w1[k * 32 + lane];
  f = fmaxf(f, 0.f);
  float o = f * w2[lane];
  for (int off = 16; off; off >>= 1) o += __shfl_xor(o, off, 32);
  if (lane == 0) out[m] = o + b2[0];
}

// ---------------------------------------------------------------------------
extern "C" void kernel_launch(void* const* d_in, const int* in_sizes, int n_in,
                              void* d_out, int out_size, void* d_ws, size_t ws_size,
                              hipStream_t stream)
{
  (void)in_sizes; (void)n_in; (void)out_size; (void)ws_size;
  const float* x  = (const float*)d_in[0];
  const int*   ei = (const int*)d_in[1];
  const float* ew = (const float*)d_in[2];
  int p = 3;
  auto F = [&](void) { return (const float*)d_in[p++]; };
  const float* emb_W = F(); const float* emb_b = F();
  struct Blk { const float *tc1_w,*tc1_b,*gcn_W,*gcn_b,*ln_g,*ln_b,*tc2_w,*tc2_b,*bn_g,*bn_b; } blk[3];
  for (int i = 0; i < 3; ++i) {
    blk[i].tc1_w = F(); blk[i].tc1_b = F(); blk[i].gcn_W = F(); blk[i].gcn_b = F();
    blk[i].ln_g  = F(); blk[i].ln_b  = F(); blk[i].tc2_w = F(); blk[i].tc2_b = F();
    blk[i].bn_g  = F(); blk[i].bn_b  = F();
  }
  const float *a_in_w = F(), *a_in_b = F(), *a_out_w = F(), *a_out_b = F(),
              *a_ln_g = F(), *a_ln_b = F();
  struct Gru { const float *Wih,*Whh,*bih,*bhh; } gru[2];
  for (int i = 0; i < 2; ++i) { gru[i].Wih = F(); gru[i].Whh = F(); gru[i].bih = F(); gru[i].bhh = F(); }
  const float *fc1_W = F(), *fc1_b = F(), *fc2_W = F(), *fc2_b = F();

  // workspace layout (~328 MB of f32)
  float* w = (float*)d_ws;
  const size_t BIG = (size_t)ROWS_ * H_;
  float* buf0 = w; w += BIG;                 // h
  float* buf1 = w; w += BIG;                 // temp / hn
  float* buf2 = w; w += BIG;                 // temp / gru ys
  float* deg  = w; w += N_;
  float* dinv = w; w += N_;
  float* nrm  = w; w += E_;
  float* stats= w; w += 128;
  float* gi   = w; w += (size_t)M_ * 192;
  float* gh   = w; w += (size_t)M_ * 192;
  float* ha   = w; w += (size_t)M_ * H_;
  float* hb   = w; w += (size_t)M_ * H_;

  // graph normalization (deg includes +1 self-loops)
  fill_kernel<<<(N_ + 255) / 256, 256, 0, stream>>>(deg, 1.0f, N_);
  deg_accum_kernel<<<(E_ + 255) / 256, 256, 0, stream>>>(ei, ew, deg);
  dinv_kernel<<<(N_ + 255) / 256, 256, 0, stream>>>(deg, dinv);
  norm_kernel<<<(E_ + 255) / 256, 256, 0, stream>>>(ei, ew, dinv, nrm);

  // embedding: (384000 x 16) @ (16 x 64) + b
  gemm_kernel<<<dim3(ROWS_ / 64, 1), 128, 0, stream>>>(
      x, 0, FIN_, emb_W, H_, 1, emb_b, buf0, 0, H_, FIN_, 0);

  for (int i = 0; i < 3; ++i) {
    conv1x3_kernel<<<dim3(B_ * T_, N_ / 16), 128, 0, stream>>>(buf0, blk[i].tc1_w, blk[i].tc1_b, buf1, 1);
    // xl = y @ gcn_W
    gemm_kernel<<<dim3(ROWS_ / 64, 1), 128, 0, stream>>>(
        buf1, 0, H_, blk[i].gcn_W, H_, 1, nullptr, buf2, 0, H_, H_, 0);
    fill_bias_kernel<<<(int)((BIG + 255) / 256), 256, 0, stream>>>(buf1, blk[i].gcn_b, BIG);
    gcn_scatter_kernel<<<dim3(E_ + N_, B_ * T_), H_, 0, stream>>>(buf2, ei, nrm, dinv, buf1);
    ln_kernel<<<ROWS_ / 8, 256, 0, stream>>>(buf1, blk[i].ln_g, blk[i].ln_b, buf2, 1, ROWS_);
    conv1x3_kernel<<<dim3(B_ * T_, N_ / 16), 128, 0, stream>>>(buf2, blk[i].tc2_w, blk[i].tc2_b, buf1, 0);
    fill_kernel<<<1, 128, 0, stream>>>(stats, 0.f, 128);
    bn_stats_kernel<<<1024, 256, 0, stream>>>(buf1, buf0, stats);
    bn_apply_kernel<<<(int)((BIG + 255) / 256), 256, 0, stream>>>(buf1, buf0, stats, blk[i].bn_g, blk[i].bn_b);
  }

  // MHA + residual + LN -> buf1 in (B*N, T, H)
  mha_kernel<<<M_, 128, 0, stream>>>(buf0, a_in_w, a_in_b, a_out_w, a_out_b, a_ln_g, a_ln_b, buf1);

  // two GRU layers, 24 sequential steps each
  const float* src = buf1;
  float* ys = buf2;
  float* last = ha;
  for (int L = 0; L < 2; ++L) {
    float* pa = ha; float* pb = hb;
    fill_kernel<<<(int)(((size_t)M_ * H_ + 255) / 256), 256, 0, stream>>>(pa, 0.f, (size_t)M_ * H_);
    for (int t = 0; t < T_; ++t) {
      // gi = x_t @ Wih^T + bih   (16000 x 64) @ (64 x 192)
      gemm_kernel<<<dim3(M_ / 64, 3), 128, 0, stream>>>(
          src, (long long)t * H_, T_ * H_, gru[L].Wih, 1, H_, gru[L].bih, gi, 0, 192, H_, 0);
      // gh = h @ Whh^T + bhh
      gemm_kernel<<<dim3(M_ / 64, 3), 128, 0, stream>>>(
          pa, 0, H_, gru[L].Whh, 1, H_, gru[L].bhh, gh, 0, 192, H_, 0);
      gru_gate_kernel<<<(int)(((size_t)M_ * H_ + 255) / 256), 256, 0, stream>>>(
          gi, gh, pa, pb, (L == 0) ? ys : nullptr, (long long)t * H_, T_ * H_);
      float* tmp = pa; pa = pb; pb = tmp;
    }
    last = pa;
    src = ys;  // layer 2 consumes layer 1's full output sequence
  }

  fc_kernel<<<(M_ + 7) / 8, 256, 0, stream>>>(last, fc1_W, fc1_b, fc2_W, fc2_b, (float*)d_out);
}